// MambaBranch_18777597018879
// MI455X (gfx1250) — compile-verified
//
#include <hip/hip_runtime.h>
#include <hip/hip_bf16.h>

#define BATCH   16
#define SEQ     1024
#define NTOK    (BATCH * SEQ)          // 16384 tokens
#define D_INPUT 64
#define D_MODEL 128
#define D_STATE 16
#define D_CONV  4
#define D_INNER 256
#define N_LAYERS 2
#define LN_EPS  1e-5f

typedef float v2f __attribute__((ext_vector_type(2)));
typedef float v8f __attribute__((ext_vector_type(8)));

__device__ __forceinline__ float silu_f(float x) { return x / (1.0f + __expf(-x)); }
__device__ __forceinline__ float softplus_f(float x) {
    return fmaxf(x, 0.0f) + log1pf(__expf(-fabsf(x)));   // max(x,0)+log1p(exp(-|x|))
}

// ---------------------------------------------------------------------------
// fp32 WMMA GEMM, compile-time N/K:  C[M,N] = A[M,K] @ B[K,N], row-major.
// One wave computes a 16 x (16*NT) tile: shared A fragment, NT independent
// accumulator chains of V_WMMA_F32_16X16X4_F32.
// MODE: 0 = plain, 1 = +bias[N], 2 = softplus(x+bias[N]), 3 = +res (residual)
// ---------------------------------------------------------------------------
template <int N, int K, int NT, int MODE>
__global__ void gemm_wmma(const float* __restrict__ A,
                          const float* __restrict__ B,
                          float* __restrict__ C,
                          const float* __restrict__ bias,
                          const float* __restrict__ res,
                          int M)
{
    constexpr int WTN = (N / 16) / NT;           // wave-tiles along N
    const int wave = (blockIdx.x * blockDim.x + threadIdx.x) >> 5;
    const int lane = threadIdx.x & 31;
    const int nTiles = (M >> 4) * WTN;
    if (wave >= nTiles) return;

    const int tm  = wave / WTN;
    const int tn0 = (wave % WTN) * NT;           // first 16-wide N tile
    const int m   = lane & 15;                   // A row / B,C col within tile
    const int hi  = lane >> 4;                   // upper-half-lane selector

    // A fragment (16x4): lane(m,hi) holds A[m][k + 2*hi + {0,1}]
    const float* Arow = A + (size_t)(tm * 16 + m) * K + 2 * hi;
    // B fragment (4x16): lane(n,hi) holds B[k + 2*hi + {0,1}][n]
    const float* Bbase = B + tn0 * 16 + m;

    v8f acc[NT];
    #pragma unroll
    for (int t = 0; t < NT; ++t) acc[t] = (v8f){0.f,0.f,0.f,0.f,0.f,0.f,0.f,0.f};

    #pragma unroll 4
    for (int k = 0; k < K; k += 4) {
        const v2f a = *(const v2f*)(Arow + k);
        const float* Bk = Bbase + (size_t)(k + 2 * hi) * N;
        #pragma unroll
        for (int t = 0; t < NT; ++t) {
            v2f b;
            b.x = Bk[t * 16];
            b.y = Bk[t * 16 + N];
            acc[t] = __builtin_amdgcn_wmma_f32_16x16x4_f32(
                false, a, false, b, (short)0, acc[t], false, false);
        }
    }

    // C/D layout: lane(n,hi), VGPR v holds C[v + 8*hi][n]
    const int n = lane & 15;
    #pragma unroll
    for (int t = 0; t < NT; ++t) {
        const int   col = (tn0 + t) * 16 + n;
        const float bv  = (MODE == 1 || MODE == 2) ? bias[col] : 0.0f;
        #pragma unroll
        for (int v = 0; v < 8; ++v) {
            const int    row = tm * 16 + v + 8 * hi;
            const size_t idx = (size_t)row * N + col;
            float val = acc[t][v] + bv;
            if (MODE == 2) val = softplus_f(val);
            if (MODE == 3) val += res[idx];
            C[idx] = val;
        }
    }
}

template <int N, int K, int NT, int MODE>
static inline void launch_gemm(const float* A, const float* B, float* C,
                               const float* bias, const float* res,
                               int M, hipStream_t stream)
{
    const int tiles  = (M / 16) * ((N / 16) / NT);
    const int blocks = (tiles + 7) / 8;          // 8 waves (256 thr) per block
    gemm_wmma<N, K, NT, MODE><<<blocks, 256, 0, stream>>>(A, B, C, bias, res, M);
}

// ---------------------------------------------------------------------------
// LayerNorm over last dim (=128). One wave32 per token, 4 elems per lane.
// ---------------------------------------------------------------------------
__global__ void layernorm_k(const float* __restrict__ x,
                            const float* __restrict__ w,
                            const float* __restrict__ b,
                            float* __restrict__ out, int ntok)
{
    const int wave = (blockIdx.x * blockDim.x + threadIdx.x) >> 5;
    const int lane = threadIdx.x & 31;
    if (wave >= ntok) return;

    const float4 v = *(const float4*)(x + (size_t)wave * D_MODEL + lane * 4);
    float s  = v.x + v.y + v.z + v.w;
    float sq = v.x * v.x + v.y * v.y + v.z * v.z + v.w * v.w;
    #pragma unroll
    for (int off = 16; off >= 1; off >>= 1) {
        s  += __shfl_xor(s,  off, 32);
        sq += __shfl_xor(sq, off, 32);
    }
    const float mu  = s * (1.0f / D_MODEL);
    const float var = sq * (1.0f / D_MODEL) - mu * mu;
    const float rs  = rsqrtf(var + LN_EPS);

    const float4 wv = *(const float4*)(w + lane * 4);
    const float4 bv = *(const float4*)(b + lane * 4);
    float4 o;
    o.x = (v.x - mu) * rs * wv.x + bv.x;
    o.y = (v.y - mu) * rs * wv.y + bv.y;
    o.z = (v.z - mu) * rs * wv.z + bv.z;
    o.w = (v.w - mu) * rs * wv.w + bv.w;
    *(float4*)(out + (size_t)wave * D_MODEL + lane * 4) = o;
}

// ---------------------------------------------------------------------------
// Causal depthwise conv1d (width 4) + bias + SiLU.
// xi = first D_INNER cols of xz (row stride 2*D_INNER).
// ---------------------------------------------------------------------------
__global__ void conv_silu_k(const float* __restrict__ xz,
                            const float* __restrict__ w,   // [D_INNER, D_CONV]
                            const float* __restrict__ cb,  // [D_INNER]
                            float* __restrict__ xc)        // [NTOK, D_INNER]
{
    const int idx = blockIdx.x * blockDim.x + threadIdx.x;
    if (idx >= NTOK * D_INNER) return;
    const int d = idx & (D_INNER - 1);
    const int t = idx >> 8;           // b*SEQ + l
    const int l = t & (SEQ - 1);

    float acc = cb[d];
    #pragma unroll
    for (int j = 0; j < D_CONV; ++j) {
        const int ll = l - (D_CONV - 1) + j;
        if (ll >= 0)
            acc += w[d * D_CONV + j] *
                   xz[(size_t)(t - (D_CONV - 1) + j) * (2 * D_INNER) + d];
    }
    xc[(size_t)t * D_INNER + d] = silu_f(acc);
}

// ---------------------------------------------------------------------------
// Selective scan: one block per batch, one thread per inner channel,
// state s[16] in registers, per-step B vector broadcast via LDS.
// ---------------------------------------------------------------------------
__global__ void scan_k(const float* __restrict__ xc,     // [NTOK, D_INNER]
                       const float* __restrict__ delta,  // [NTOK, D_INNER]
                       const float* __restrict__ xdbl,   // [NTOK, 2*D_STATE]
                       const float* __restrict__ xz,     // [NTOK, 2*D_INNER]
                       const float* __restrict__ A_log,  // [D_INNER, D_STATE]
                       const float* __restrict__ Dp,     // [D_INNER]
                       float* __restrict__ y)            // [NTOK, D_INNER]
{
    const int b = blockIdx.x;
    const int d = threadIdx.x;  // 0..255

    float Arow[D_STATE];
    #pragma unroll
    for (int n = 0; n < D_STATE; ++n)
        Arow[n] = -__expf(A_log[d * D_STATE + n]);
    const float Dd = Dp[d];

    float s[D_STATE];
    #pragma unroll
    for (int n = 0; n < D_STATE; ++n) s[n] = 0.0f;

    __shared__ float Bsh[D_STATE];

    for (int l = 0; l < SEQ; ++l) {
        const int t = b * SEQ + l;
        if (d < D_STATE)
            Bsh[d] = xdbl[(size_t)t * (2 * D_STATE) + D_STATE + d];
        __syncthreads();

        const float dlt = delta[(size_t)t * D_INNER + d];
        const float xv  = xc[(size_t)t * D_INNER + d];
        float acc = 0.0f;
        #pragma unroll
        for (int n = 0; n < D_STATE; ++n) {
            const float dA = __expf(dlt * Arow[n]);
            s[n] = dA * s[n] + (dlt * Bsh[n]) * xv;
            acc += s[n] * Bsh[n];
        }
        acc += Dd * xv;
        const float zv = xz[(size_t)t * (2 * D_INNER) + D_INNER + d];
        acc *= silu_f(zv);
        y[(size_t)t * D_INNER + d] = acc;
        __syncthreads();
    }
}

// ---------------------------------------------------------------------------
extern "C" void kernel_launch(void* const* d_in, const int* in_sizes, int n_in,
                              void* d_out, int out_size, void* d_ws, size_t ws_size,
                              hipStream_t stream)
{
    const float* x       = (const float*)d_in[0];
    const float* inp_w   = (const float*)d_in[1];
    const float* inp_b   = (const float*)d_in[2];
    const float* ln_w    = (const float*)d_in[3];
    const float* ln_b    = (const float*)d_in[4];
    const float* in_w    = (const float*)d_in[5];
    const float* conv_w  = (const float*)d_in[6];
    const float* conv_b  = (const float*)d_in[7];
    const float* xproj_w = (const float*)d_in[8];
    const float* dt_w    = (const float*)d_in[9];
    const float* dt_b    = (const float*)d_in[10];
    const float* A_log   = (const float*)d_in[11];
    const float* Dp      = (const float*)d_in[12];
    const float* out_w   = (const float*)d_in[13];
    const float* fn_w    = (const float*)d_in[14];
    const float* fn_b    = (const float*)d_in[15];

    float* ws = (float*)d_ws;
    float* h  = ws;                              // NTOK * D_MODEL
    float* hn = h  + (size_t)NTOK * D_MODEL;     // NTOK * D_MODEL
    float* xz = hn + (size_t)NTOK * D_MODEL;     // NTOK * 2*D_INNER
    float* xc = xz + (size_t)NTOK * 2 * D_INNER; // NTOK * D_INNER
    float* dl = xc + (size_t)NTOK * D_INNER;     // NTOK * D_INNER
    float* xd = dl + (size_t)NTOK * D_INNER;     // NTOK * 2*D_STATE
    float* yy = xd + (size_t)NTOK * 2 * D_STATE; // NTOK * D_INNER

    // h = x @ inp_w + inp_b                       (K=64,  N=128)
    launch_gemm<D_MODEL, D_INPUT, 4, 1>(x, inp_w, h, inp_b, nullptr, NTOK, stream);

    for (int i = 0; i < N_LAYERS; ++i) {
        // hn = LN(h)
        layernorm_k<<<NTOK / 8, 256, 0, stream>>>(h, ln_w + i * D_MODEL,
                                                  ln_b + i * D_MODEL, hn, NTOK);
        // xz = hn @ in_w[i]                       (K=128, N=512)
        launch_gemm<2 * D_INNER, D_MODEL, 4, 0>(
            hn, in_w + (size_t)i * D_MODEL * 2 * D_INNER, xz,
            nullptr, nullptr, NTOK, stream);
        // xc = silu(causal_dwconv(xi) + conv_b)
        conv_silu_k<<<(NTOK * D_INNER) / 256, 256, 0, stream>>>(
            xz, conv_w + (size_t)i * D_INNER * D_CONV, conv_b + i * D_INNER, xc);
        // xd = xc @ xproj_w[i]                    (K=256, N=32)
        launch_gemm<2 * D_STATE, D_INNER, 2, 0>(
            xc, xproj_w + (size_t)i * D_INNER * 2 * D_STATE, xd,
            nullptr, nullptr, NTOK, stream);
        // dl = softplus(xc @ dt_w[i] + dt_b[i])   (K=256, N=256)
        launch_gemm<D_INNER, D_INNER, 4, 2>(
            xc, dt_w + (size_t)i * D_INNER * D_INNER, dl,
            dt_b + i * D_INNER, nullptr, NTOK, stream);
        // yy = selective scan
        scan_k<<<BATCH, D_INNER, 0, stream>>>(
            xc, dl, xd, xz, A_log + (size_t)i * D_INNER * D_STATE,
            Dp + i * D_INNER, yy);
        // h = h + yy @ out_w[i]                   (K=256, N=128)
        launch_gemm<D_MODEL, D_INNER, 4, 3>(
            yy, out_w + (size_t)i * D_INNER * D_MODEL, h,
            nullptr, h, NTOK, stream);
    }

    // d_out = LN(h, fn_w, fn_b)
    layernorm_k<<<NTOK / 8, 256, 0, stream>>>(h, fn_w, fn_b, (float*)d_out, NTOK);
}